// GNN_59158879535179
// MI455X (gfx1250) — compile-verified
//
#include <hip/hip_runtime.h>
#include <math.h>

typedef __attribute__((ext_vector_type(2))) float v2f;
typedef __attribute__((ext_vector_type(8))) float v8f;

#define TPB 256

// ---------------------------------------------------------------------------
// 1) deg[i] = 1.0f (self-loop), zero the 64-graph pooling accumulators
// ---------------------------------------------------------------------------
__global__ void gnn_init_kernel(float* __restrict__ deg, float* __restrict__ sums,
                                float* __restrict__ cnts, int N) {
  int i = blockIdx.x * TPB + threadIdx.x;
  if (i < N) deg[i] = 1.0f;
  if (blockIdx.x == 0 && threadIdx.x < 64) {
    sums[threadIdx.x] = 0.0f;
    cnts[threadIdx.x] = 0.0f;
  }
}

// ---------------------------------------------------------------------------
// 2) in-degree over dst
// ---------------------------------------------------------------------------
__global__ void gnn_degree_kernel(const long long* __restrict__ dst,
                                  float* __restrict__ deg, int E) {
  int e = blockIdx.x * TPB + threadIdx.x;
  if (e < E) atomicAdd(&deg[(int)dst[e]], 1.0f);
}

// ---------------------------------------------------------------------------
// 3) dinv = rsqrt(deg) in place (deg >= 1 always, matches reference)
// ---------------------------------------------------------------------------
__global__ void gnn_dinv_kernel(float* __restrict__ deg, int N) {
  int i = blockIdx.x * TPB + threadIdx.x;
  if (i < N) deg[i] = rsqrtf(deg[i]);
}

// ---------------------------------------------------------------------------
// 4) WMMA node transform: hs[i,:] = (x[i,:] @ W1) * dinv[i]; agg seeded = hs
//    One wave32 computes a 16-node x 16-feature tile with two
//    V_WMMA_F32_16X16X4_F32 ops (K = 0..3 and K = 4..6, K=7 padded with 0).
//    A layout: lanes 0-15 hold M=lane, K={0,1}; lanes 16-31 hold K={2,3}.
//    B layout: VGPR0 holds rows K=kb (lanes 0-15 N=0..15, lanes 16-31 K=kb+2).
//    D layout: VGPR r = row r (lanes 0-15) / row r+8 (lanes 16-31), N = lane&15.
// ---------------------------------------------------------------------------
__global__ void gnn_transform1_wmma(const float* __restrict__ x,
                                    const float* __restrict__ W1,
                                    const float* __restrict__ dinv,
                                    float* __restrict__ hs,
                                    float* __restrict__ agg,
                                    int nTiles) {
  int wave = (blockIdx.x * TPB + threadIdx.x) >> 5;
  int lane = threadIdx.x & 31;
  if (wave >= nTiles) return;          // wave-uniform: EXEC stays all-ones

  const int half = lane >> 4;          // 0: K pair {0,1}/{4,5}; 1: {2,3}/{6,7}
  const int l15  = lane & 15;
  const int kb   = half * 2;
  const int node = wave * 16 + l15;    // A-matrix row M = lane&15

  // A tiles (16x4 each): features K=0..3 then K=4..7 (K=7 zero pad)
  const float* xrow = x + (long long)node * 7;
  v2f a0, a1;
  a0.x = xrow[kb + 0];
  a0.y = xrow[kb + 1];
  a1.x = xrow[4 + kb];                           // K=4 or 6
  a1.y = (half == 0) ? xrow[5] : 0.0f;           // K=5 valid, K=7 -> 0

  // B tiles (4x16 each): row K across lanes, W1 is [7][16] row-major
  v2f b0, b1v;
  b0.x  = W1[(kb + 0) * 16 + l15];
  b0.y  = W1[(kb + 1) * 16 + l15];
  b1v.x = W1[(4 + kb) * 16 + l15];
  b1v.y = (half == 0) ? W1[5 * 16 + l15] : 0.0f; // K=7 -> 0

  v8f c = {};
  c = __builtin_amdgcn_wmma_f32_16x16x4_f32(false, a0, false, b0,
                                            (short)0, c, false, false);
  c = __builtin_amdgcn_wmma_f32_16x16x4_f32(false, a1, false, b1v,
                                            (short)0, c, false, false);

  // Scale by dinv[row] and store; also seed agg with the self-loop term.
  const int mbase = wave * 16 + 8 * half;
#pragma unroll
  for (int r = 0; r < 8; ++r) {
    int m = mbase + r;
    float v = c[r] * dinv[m];
    hs[(long long)m * 16 + l15]  = v;
    agg[(long long)m * 16 + l15] = v;
  }
}

// Scalar tail for N % 16 != 0 (unused for N=500000 but kept for generality)
__global__ void gnn_transform1_tail(const float* __restrict__ x,
                                    const float* __restrict__ W1,
                                    const float* __restrict__ dinv,
                                    float* __restrict__ hs,
                                    float* __restrict__ agg,
                                    int start, int N) {
  int i = start + blockIdx.x * TPB + threadIdx.x;
  if (i >= N) return;
  float di = dinv[i];
  float xr[7];
#pragma unroll
  for (int k = 0; k < 7; ++k) xr[k] = x[(long long)i * 7 + k];
#pragma unroll
  for (int f = 0; f < 16; ++f) {
    float a = 0.0f;
#pragma unroll
    for (int k = 0; k < 7; ++k) a += xr[k] * W1[k * 16 + f];
    a *= di;
    hs[(long long)i * 16 + f]  = a;
    agg[(long long)i * 16 + f] = a;
  }
}

// ---------------------------------------------------------------------------
// 5) edge pass layer 1: thread = (edge, feature); 16 lanes cover one edge's
//    feature row -> 64B coalesced gather + 64B coalesced atomic scatter-add
// ---------------------------------------------------------------------------
__global__ void gnn_edge1_kernel(const long long* __restrict__ src,
                                 const long long* __restrict__ dst,
                                 const float* __restrict__ hs,
                                 float* __restrict__ agg, long long E) {
  long long t = (long long)blockIdx.x * TPB + threadIdx.x;
  long long e = t >> 4;
  if (e >= E) return;
  int f = (int)(t & 15);
  long long s = src[e];
  long long d = dst[e];
  atomicAdd(&agg[d * 16 + f], hs[s * 16 + f]);
}

// ---------------------------------------------------------------------------
// 6) node pass: h1 = relu(dinv*agg + b1); t = (h1 . W2) * dinv; agg2 seeded
// ---------------------------------------------------------------------------
__global__ void gnn_node1_kernel(const float* __restrict__ agg,
                                 const float* __restrict__ dinv,
                                 const float* __restrict__ b1,
                                 const float* __restrict__ W2,
                                 float* __restrict__ tbuf,
                                 float* __restrict__ agg2, int N) {
  int i = blockIdx.x * TPB + threadIdx.x;
  if (i >= N) return;
  float di = dinv[i];
  const float4* a4 = (const float4*)(agg + (long long)i * 16);
  float acc = 0.0f;
#pragma unroll
  for (int q = 0; q < 4; ++q) {
    float4 v = a4[q];
    float h0 = fmaxf(di * v.x + b1[q * 4 + 0], 0.0f);
    float h1 = fmaxf(di * v.y + b1[q * 4 + 1], 0.0f);
    float h2 = fmaxf(di * v.z + b1[q * 4 + 2], 0.0f);
    float h3 = fmaxf(di * v.w + b1[q * 4 + 3], 0.0f);
    acc += h0 * W2[q * 4 + 0] + h1 * W2[q * 4 + 1] +
           h2 * W2[q * 4 + 2] + h3 * W2[q * 4 + 3];
  }
  float tv = acc * di;   // pre-scale by dinv[src] for layer-2 propagation
  tbuf[i] = tv;
  agg2[i] = tv;          // self-loop seed
}

// ---------------------------------------------------------------------------
// 7) edge pass layer 2: one float atomic per edge
// ---------------------------------------------------------------------------
__global__ void gnn_edge2_kernel(const long long* __restrict__ src,
                                 const long long* __restrict__ dst,
                                 const float* __restrict__ tbuf,
                                 float* __restrict__ agg2, int E) {
  int e = blockIdx.x * TPB + threadIdx.x;
  if (e >= E) return;
  atomicAdd(&agg2[dst[e]], tbuf[src[e]]);
}

// ---------------------------------------------------------------------------
// 8) node pass 2 + global mean pool (LDS-staged, batch is sorted -> low
//    contention; 2 global atomics per graph present in the block)
// ---------------------------------------------------------------------------
__global__ void gnn_node2_pool(const float* __restrict__ agg2,
                               const float* __restrict__ dinv,
                               const float* __restrict__ b2,
                               const long long* __restrict__ batch,
                               float* __restrict__ sums,
                               float* __restrict__ cnts, int N) {
  __shared__ float ls[64];
  __shared__ float lc[64];
  int tid = threadIdx.x;
  if (tid < 64) { ls[tid] = 0.0f; lc[tid] = 0.0f; }
  __syncthreads();
  int i = blockIdx.x * TPB + tid;
  if (i < N) {
    float h2 = dinv[i] * agg2[i] + b2[0];
    int b = (int)batch[i];
    atomicAdd(&ls[b], h2);
    atomicAdd(&lc[b], 1.0f);
  }
  __syncthreads();
  if (tid < 64 && lc[tid] != 0.0f) {
    atomicAdd(&sums[tid], ls[tid]);
    atomicAdd(&cnts[tid], lc[tid]);
  }
}

// ---------------------------------------------------------------------------
// 9) mean + sigmoid
// ---------------------------------------------------------------------------
__global__ void gnn_final_kernel(const float* __restrict__ sums,
                                 const float* __restrict__ cnts,
                                 float* __restrict__ out) {
  int g = threadIdx.x;
  if (g < 64) {
    float c = fmaxf(cnts[g], 1.0f);
    float v = sums[g] / c;
    out[g] = 1.0f / (1.0f + expf(-v));
  }
}

extern "C" void kernel_launch(void* const* d_in, const int* in_sizes, int n_in,
                              void* d_out, int out_size, void* d_ws, size_t ws_size,
                              hipStream_t stream) {
  const float*     x     = (const float*)d_in[0];      // [N,7]
  const float*     W1    = (const float*)d_in[1];      // [7,16]
  const float*     b1    = (const float*)d_in[2];      // [16]
  const float*     W2    = (const float*)d_in[3];      // [16,1]
  const float*     b2    = (const float*)d_in[4];      // [1]
  const long long* eidx  = (const long long*)d_in[5];  // [2,E] int64
  const long long* batch = (const long long*)d_in[6];  // [N]   int64

  const int N = in_sizes[0] / 7;
  const int E = in_sizes[5] / 2;
  const long long* src = eidx;
  const long long* dst = eidx + E;

  // Workspace carve-up (floats), 256B-aligned sections
  char* ws = (char*)d_ws;
  size_t off = 0;
  auto carve = [&](size_t nFloats) -> float* {
    float* p = (float*)(ws + off);
    off += ((nFloats * sizeof(float)) + 255) & ~(size_t)255;
    return p;
  };
  float* deg  = carve((size_t)N);        // becomes dinv in place
  float* hs   = carve((size_t)N * 16);
  float* agg  = carve((size_t)N * 16);
  float* tbuf = carve((size_t)N);
  float* agg2 = carve((size_t)N);
  float* sums = carve(64);
  float* cnts = carve(64);
  (void)ws_size;

  const int nBlocksN = (N + TPB - 1) / TPB;
  const int nBlocksE = (E + TPB - 1) / TPB;

  gnn_init_kernel<<<nBlocksN, TPB, 0, stream>>>(deg, sums, cnts, N);
  gnn_degree_kernel<<<nBlocksE, TPB, 0, stream>>>(dst, deg, E);
  gnn_dinv_kernel<<<nBlocksN, TPB, 0, stream>>>(deg, N);

  const int nTiles = N / 16;                      // full 16-node WMMA tiles
  if (nTiles > 0) {
    const int wavesPerBlock = TPB / 32;
    const int nBlocksT = (nTiles + wavesPerBlock - 1) / wavesPerBlock;
    gnn_transform1_wmma<<<nBlocksT, TPB, 0, stream>>>(x, W1, deg, hs, agg, nTiles);
  }
  const int tail = N - nTiles * 16;
  if (tail > 0) {
    gnn_transform1_tail<<<(tail + TPB - 1) / TPB, TPB, 0, stream>>>(
        x, W1, deg, hs, agg, nTiles * 16, N);
  }

  const long long ethreads = (long long)E * 16;
  const int nBlocksE16 = (int)((ethreads + TPB - 1) / TPB);
  gnn_edge1_kernel<<<nBlocksE16, TPB, 0, stream>>>(src, dst, hs, agg, (long long)E);

  gnn_node1_kernel<<<nBlocksN, TPB, 0, stream>>>(agg, deg, b1, W2, tbuf, agg2, N);
  gnn_edge2_kernel<<<nBlocksE, TPB, 0, stream>>>(src, dst, tbuf, agg2, E);
  gnn_node2_pool<<<nBlocksN, TPB, 0, stream>>>(agg2, deg, b2, batch, sums, cnts, N);
  gnn_final_kernel<<<1, 64, 0, stream>>>(sums, cnts, (float*)d_out);
}